// PEAGATRecsysModel_51118700757607
// MI455X (gfx1250) — compile-verified
//
#include <hip/hip_runtime.h>
#include <hip/hip_bf16.h>

#define N_NODES   100000
#define HID       64
#define N_CH      9
#define N_EDGES   500000
#define ET        (N_EDGES + N_NODES)   // edges + self loops
#define NEG_SLOPE 0.2f

typedef float v2f __attribute__((ext_vector_type(2)));
typedef float v8f __attribute__((ext_vector_type(8)));

// ---------------- utility: zero fill (float4 wide) ----------------
__global__ void zero4_kernel(float4* __restrict__ p, int n4) {
  int i = blockIdx.x * blockDim.x + threadIdx.x;
  if (i < n4) p[i] = make_float4(0.f, 0.f, 0.f, 0.f);
}

// ---------------- ordered-uint encoding for float atomicMax ----------------
__device__ __forceinline__ unsigned fenc(float f) {
  unsigned b = __float_as_uint(f);
  return (b & 0x80000000u) ? ~b : (b | 0x80000000u);
}
__device__ __forceinline__ float fdec(unsigned u) {
  return (u & 0x80000000u) ? __uint_as_float(u & 0x7FFFFFFFu) : __uint_as_float(~u);
}

// ---------------- GEMM: C[M,64] = A[M,64] x W[64,64] via f32 WMMA ----------------
// One wave32 per 16x16 output tile; K-loop of 16 x v_wmma_f32_16x16x4_f32.
__global__ __launch_bounds__(256) void gemm64_wmma(const float* __restrict__ A,
                                                   const float* __restrict__ W,
                                                   float* __restrict__ C, int mtiles) {
  int wave  = (blockIdx.x * blockDim.x + threadIdx.x) >> 5;
  int lane  = threadIdx.x & 31;
  int mtile = wave >> 2;        // 4 N-tiles of 16 cols
  int ntile = wave & 3;
  if (mtile >= mtiles) return;  // wave-uniform branch: EXEC stays all-1s inside
  int half = lane >> 4;         // 0: K pair {k0,k0+1}, 1: {k0+2,k0+3}
  int l15  = lane & 15;

  const float* Arow = A + (size_t)(mtile * 16 + l15) * HID;
  const float* Wc   = W + ntile * 16 + l15;

  v8f acc = {};
#pragma unroll
  for (int kk = 0; kk < 16; ++kk) {
    int k0 = kk * 4 + half * 2;
    float2 av = *reinterpret_cast<const float2*>(Arow + k0);  // 8B aligned
    v2f a; a.x = av.x; a.y = av.y;
    v2f b; b.x = Wc[(size_t)k0 * HID]; b.y = Wc[(size_t)(k0 + 1) * HID];
    acc = __builtin_amdgcn_wmma_f32_16x16x4_f32(false, a, false, b,
                                                (short)0, acc, false, false);
  }
  int row0 = mtile * 16 + half * 8;
  float* Co = C + (size_t)row0 * HID + ntile * 16 + l15;
#pragma unroll
  for (int v = 0; v < 8; ++v) Co[(size_t)v * HID] = acc[v];
}

// ---------------- per-node dots: s_src[n] = h[n].a_src, s_dst[n] = h[n].a_dst ----
__global__ __launch_bounds__(256) void node_dots(const float* __restrict__ h,
                                                 const float* __restrict__ asrc,
                                                 const float* __restrict__ adst,
                                                 float* __restrict__ ssrc,
                                                 float* __restrict__ sdst, int n) {
  int wave = (blockIdx.x * blockDim.x + threadIdx.x) >> 5;
  int lane = threadIdx.x & 31;
  if (wave >= n) return;
  const float* row = h + (size_t)wave * HID;
  float h0 = row[lane], h1 = row[lane + 32];
  float a = h0 * asrc[lane] + h1 * asrc[lane + 32];
  float b = h0 * adst[lane] + h1 * adst[lane + 32];
#pragma unroll
  for (int off = 16; off > 0; off >>= 1) {
    a += __shfl_xor(a, off, 32);
    b += __shfl_xor(b, off, 32);
  }
  if (lane == 0) { ssrc[wave] = a; sdst[wave] = b; }
}

// ---------------- edge pass 1: e = leaky_relu(s_src[src]+s_dst[dst]); segment max -
__global__ void edge_max(const int* __restrict__ src, const int* __restrict__ dst,
                         const float* __restrict__ ssrc, const float* __restrict__ sdst,
                         float* __restrict__ ebuf, unsigned* __restrict__ menc) {
  int i = blockIdx.x * blockDim.x + threadIdx.x;
  if (i >= ET) return;
  int s, d;
  if (i < N_EDGES) { s = src[i]; d = dst[i]; } else { s = d = i - N_EDGES; }
  float e = ssrc[s] + sdst[d];
  e = e > 0.f ? e : NEG_SLOPE * e;
  ebuf[i] = e;
  atomicMax(&menc[d], fenc(e));
}

// ---------------- edge pass 2: ex = exp(e - m[dst]); segment sum ------------------
__global__ void edge_expsum(const int* __restrict__ src, const int* __restrict__ dst,
                            float* __restrict__ ebuf, const unsigned* __restrict__ menc,
                            float* __restrict__ denom) {
  int i = blockIdx.x * blockDim.x + threadIdx.x;
  if (i >= ET) return;
  int d = (i < N_EDGES) ? dst[i] : (i - N_EDGES);
  float ex = __expf(ebuf[i] - fdec(menc[d]));
  ebuf[i] = ex;
  atomicAdd(&denom[d], ex);
}

// ---------------- edge pass 3: out[dst] += (ex/denom[dst]) * h[src] ---------------
// one wave32 per edge; lane handles features {lane, lane+32}
__global__ __launch_bounds__(256) void edge_scatter(const float* __restrict__ h,
                                                    const float* __restrict__ ebuf,
                                                    const float* __restrict__ denom,
                                                    const int* __restrict__ src,
                                                    const int* __restrict__ dst,
                                                    float* __restrict__ outp, int ostride) {
  int wave = (blockIdx.x * blockDim.x + threadIdx.x) >> 5;
  int lane = threadIdx.x & 31;
  if (wave >= ET) return;
  int s, d;
  if (wave < N_EDGES) { s = src[wave]; d = dst[wave]; } else { s = d = wave - N_EDGES; }
  float alpha = ebuf[wave] / denom[d];
  const float* hs = h + (size_t)s * HID;
  float* od = outp + (size_t)d * ostride;
  atomicAdd(od + lane,      alpha * hs[lane]);
  atomicAdd(od + lane + 32, alpha * hs[lane + 32]);
}

// ---------------- layer-1 finalize: x = relu(x + b) -------------------------------
__global__ void fin_relu_bias(float* __restrict__ x, const float* __restrict__ b, int total) {
  int i = blockIdx.x * blockDim.x + threadIdx.x;
  if (i >= total) return;
  float v = x[i] + b[i & (HID - 1)];
  x[i] = v > 0.f ? v : 0.f;
}

// ---------------- layer-2 finalize (strided slice of x2_all): x += b --------------
__global__ void fin_bias_strided(float* __restrict__ x2, int coff,
                                 const float* __restrict__ b, int total) {
  int i = blockIdx.x * blockDim.x + threadIdx.x;
  if (i >= total) return;
  int n = i >> 6, f = i & (HID - 1);
  x2[(size_t)n * (N_CH * HID) + coff + f] += b[f];
}

// ---------------- final channel attention: wave per node --------------------------
__global__ __launch_bounds__(256) void final_attn(const float* __restrict__ x2,
                                                  const float* __restrict__ att,
                                                  float* __restrict__ out, int n) {
  int wave = (blockIdx.x * blockDim.x + threadIdx.x) >> 5;
  int lane = threadIdx.x & 31;
  if (wave >= n) return;
  const float* xr = x2 + (size_t)wave * (N_CH * HID);
  float xa[N_CH], xb[N_CH], sc[N_CH];
#pragma unroll
  for (int c = 0; c < N_CH; ++c) {
    xa[c] = xr[c * HID + lane];
    xb[c] = xr[c * HID + lane + 32];
    float s = xa[c] * att[c * HID + lane] + xb[c] * att[c * HID + lane + 32];
#pragma unroll
    for (int off = 16; off > 0; off >>= 1) s += __shfl_xor(s, off, 32);
    sc[c] = s;   // broadcast to all lanes via full xor-reduction
  }
  float mx = sc[0];
#pragma unroll
  for (int c = 1; c < N_CH; ++c) mx = fmaxf(mx, sc[c]);
  float sum = 0.f;
#pragma unroll
  for (int c = 0; c < N_CH; ++c) { float w = __expf(sc[c] - mx); sc[c] = w; sum += w; }
  float inv = 1.f / sum;
  float oa = 0.f, ob = 0.f;
#pragma unroll
  for (int c = 0; c < N_CH; ++c) { float w = sc[c] * inv; oa += w * xa[c]; ob += w * xb[c]; }
  out[(size_t)wave * HID + lane]      = oa;
  out[(size_t)wave * HID + lane + 32] = ob;
}

// =================================================================================
extern "C" void kernel_launch(void* const* d_in, const int* in_sizes, int n_in,
                              void* d_out, int out_size, void* d_ws, size_t ws_size,
                              hipStream_t stream) {
  const float* emb   = (const float*)d_in[0];   // [N,64]
  const float* W1    = (const float*)d_in[1];   // [9,64,64]
  const float* as1   = (const float*)d_in[2];   // [9,64]
  const float* ad1   = (const float*)d_in[3];
  const float* b1    = (const float*)d_in[4];
  const float* W2    = (const float*)d_in[5];   // [9,64,64]
  const float* as2   = (const float*)d_in[6];
  const float* ad2   = (const float*)d_in[7];
  const float* b2    = (const float*)d_in[8];
  const float* att   = (const float*)d_in[9];   // [9,64]
  const int*   eidx  = (const int*)d_in[10];    // [9,2,2,E]
  float* out = (float*)d_out;

  // workspace carve-up (all counts are multiples of 4 floats)
  char* ws = (char*)d_ws;
  float*    h     = (float*)ws;                       ws += (size_t)N_NODES * HID * 4;          // 25.6 MB
  float*    x1    = (float*)ws;                       ws += (size_t)N_NODES * HID * 4;          // 25.6 MB
  float*    x2    = (float*)ws;                       ws += (size_t)N_NODES * N_CH * HID * 4;   // 230.4 MB
  float*    ssrc  = (float*)ws;                       ws += (size_t)N_NODES * 4;
  float*    sdst  = (float*)ws;                       ws += (size_t)N_NODES * 4;
  unsigned* menc  = (unsigned*)ws;                    ws += (size_t)N_NODES * 4;
  float*    denom = (float*)ws;                       ws += (size_t)N_NODES * 4;
  float*    ebuf  = (float*)ws;                       ws += (size_t)ET * 4;

  const int ZB = 256;
  const int NF  = N_NODES * HID;                 // 6.4M
  const int gemm_blocks  = ((N_NODES / 16) * 4 * 32) / 256;        // 3125
  const int dots_blocks  = (N_NODES * 32 + 255) / 256;             // 12500
  const int edge_blocks  = (ET + 255) / 256;                       // 2344
  const int scat_blocks  = (ET * 32 + 255) / 256;                  // 75000
  const int fin_blocks   = (NF + 255) / 256;                       // 25000

  // zero the big channel-stacked output once
  zero4_kernel<<<((size_t)N_NODES * N_CH * HID / 4 + ZB - 1) / ZB, ZB, 0, stream>>>(
      (float4*)x2, N_NODES * N_CH * HID / 4);

  for (int c = 0; c < N_CH; ++c) {
    for (int layer = 0; layer < 2; ++layer) {
      const float* X  = layer ? x1 : emb;
      const float* W  = (layer ? W2  : W1 ) + (size_t)c * HID * HID;
      const float* as = (layer ? as2 : as1) + (size_t)c * HID;
      const float* ad = (layer ? ad2 : ad1) + (size_t)c * HID;
      const int*   srcp = eidx + (((size_t)c * 2 + layer) * 2 + 0) * N_EDGES;
      const int*   dstp = srcp + N_EDGES;
      float* O    = layer ? (x2 + (size_t)c * HID) : x1;
      int ostride = layer ? (N_CH * HID) : HID;

      if (layer == 0)
        zero4_kernel<<<(NF / 4 + ZB - 1) / ZB, ZB, 0, stream>>>((float4*)x1, NF / 4);
      zero4_kernel<<<(N_NODES / 4 + ZB - 1) / ZB, ZB, 0, stream>>>((float4*)menc, N_NODES / 4);
      zero4_kernel<<<(N_NODES / 4 + ZB - 1) / ZB, ZB, 0, stream>>>((float4*)denom, N_NODES / 4);

      gemm64_wmma<<<gemm_blocks, 256, 0, stream>>>(X, W, h, N_NODES / 16);
      node_dots<<<dots_blocks, 256, 0, stream>>>(h, as, ad, ssrc, sdst, N_NODES);
      edge_max<<<edge_blocks, 256, 0, stream>>>(srcp, dstp, ssrc, sdst, ebuf, menc);
      edge_expsum<<<edge_blocks, 256, 0, stream>>>(srcp, dstp, ebuf, menc, denom);
      edge_scatter<<<scat_blocks, 256, 0, stream>>>(h, ebuf, denom, srcp, dstp, O, ostride);

      if (layer == 0)
        fin_relu_bias<<<fin_blocks, 256, 0, stream>>>(x1, b1 + (size_t)c * HID, NF);
      else
        fin_bias_strided<<<fin_blocks, 256, 0, stream>>>(x2, c * HID, b2 + (size_t)c * HID, NF);
    }
  }

  final_attn<<<dots_blocks, 256, 0, stream>>>(x2, att, out, N_NODES);
}